// EpisodicMemory_53042846106048
// MI455X (gfx1250) — compile-verified
//
#include <hip/hip_runtime.h>
#include <math.h>

// EpisodicMemory sliding-window attention for MI455X (gfx1250, wave32).
// Pipeline: [zero attn 268MB] -> [Q/K projections: async-to-LDS double-buffered
//           V_WMMA_F32_16X16X4_F32 GEMM] -> [banded scores + softmax + scatter + P@V].

typedef __attribute__((ext_vector_type(2))) float v2f;
typedef __attribute__((ext_vector_type(4))) float v4f;
typedef __attribute__((ext_vector_type(8))) float v8f;

#define NB    4          // batch
#define SEQ   4096       // sequence length
#define D2    512        // 2*DIM
#define WIN   64         // sliding window (strictly past)
#define MROWS (NB * SEQ) // 16384 flattened rows
#define BAND  80         // 64 window + 16 query block span
#define SCALE 0.04419417382415922f   // 1/sqrt(512)

static __device__ __forceinline__ v8f wmma_f32(v2f a, v2f b, v8f c) {
  // D = A(16x4,f32) * B(4x16,f32) + C(16x16,f32); 8-arg VOP3P form.
  return __builtin_amdgcn_wmma_f32_16x16x4_f32(
      /*neg_a=*/false, a, /*neg_b=*/false, b,
      /*c_mod=*/(short)0, c, /*reuse_a=*/false, /*reuse_b=*/false);
}

// Generic LDS pointer -> 32-bit LDS byte offset (aperture form keeps it in low bits).
static __device__ __forceinline__ unsigned lds_off(const void* p) {
  return (unsigned)(size_t)p;
}

// gfx1250 async copy: 16B global -> LDS, tracked by ASYNCcnt (no VGPR round-trip).
static __device__ __forceinline__ void async_copy_b128(const float* g, unsigned lo) {
  asm volatile("global_load_async_to_lds_b128 %0, %1, off"
               :: "v"(lo), "v"(g) : "memory");
}

// ---------------------------------------------------------------------------
// Kernel 1: zero the dense attn output (harness poisons d_out with 0xAA).
// ---------------------------------------------------------------------------
__global__ void zero_attn_kernel(v4f* __restrict__ p, size_t n4) {
  size_t i      = (size_t)blockIdx.x * blockDim.x + threadIdx.x;
  size_t stride = (size_t)gridDim.x * blockDim.x;
  v4f z = {0.0f, 0.0f, 0.0f, 0.0f};
  for (; i < n4; i += stride) p[i] = z;
}

// ---------------------------------------------------------------------------
// Kernel 2: Y[M,512] = X[M,512] @ W^T  (torch nn.Linear convention).
// 64x64 output tile per block, 4 waves, each wave = 16x64 stripe (4 WMMA tiles).
// K is consumed in KC=32 chunks, double-buffered via async-to-LDS.
// ---------------------------------------------------------------------------
#define KC 32
#define TP 36  // LDS pitch for 32-wide fp32 tiles (mult of 4; conflict-free rows)

__global__ __launch_bounds__(128)
void proj_kernel(const float* __restrict__ X, const float* __restrict__ Wq,
                 const float* __restrict__ Wk, float* __restrict__ Qo,
                 float* __restrict__ Ko) {
  __shared__ float At[2][64 * TP];  // X tile   [m 0..63][k 0..31]
  __shared__ float Bt[2][64 * TP];  // W tile   [n 0..63][k 0..31]

  const int tid  = threadIdx.x;
  const int lane = tid & 31;
  const int wave = tid >> 5;
  const int half = lane >> 4;    // 0: K=k0,k0+1  1: K=k0+2,k0+3
  const int l15  = lane & 15;
  const int n0   = blockIdx.x * 64;
  const int m0   = blockIdx.y * 64;

  const float* Wm = (blockIdx.z == 0) ? Wq : Wk;
  float*       Y  = (blockIdx.z == 0) ? Qo : Ko;

  v8f acc[4];
#pragma unroll
  for (int nt = 0; nt < 4; ++nt) acc[nt] = (v8f)0.0f;

  // Issue async stage of a 64x32 X tile + 64x32 W tile (8 b128 ops / thread).
  auto stage = [&](int buf, int kc) {
    for (int idx = tid; idx < 64 * 8; idx += 128) {
      int r  = idx >> 3;
      int c4 = (idx & 7) << 2;
      async_copy_b128(&X[(size_t)(m0 + r) * D2 + kc + c4],
                      lds_off(&At[buf][r * TP + c4]));
      async_copy_b128(&Wm[(size_t)(n0 + r) * D2 + kc + c4],
                      lds_off(&Bt[buf][r * TP + c4]));
    }
  };

  stage(0, 0);
  int buf = 0;
  for (int kc = 0; kc < D2; kc += KC) {
    if (kc + KC < D2) {
      stage(buf ^ 1, kc + KC);
      // Async loads retire in order: <=8 outstanding => current buffer done,
      // next buffer still in flight (overlap with compute below).
      asm volatile("s_wait_asynccnt 0x8" ::: "memory");
    } else {
      asm volatile("s_wait_asynccnt 0x0" ::: "memory");
    }
    __syncthreads();

    const float* arow = &At[buf][(wave * 16 + l15) * TP + (half << 1)];
#pragma unroll
    for (int k0 = 0; k0 < KC; k0 += 4) {
      v2f a = *(const v2f*)&arow[k0];
#pragma unroll
      for (int nt = 0; nt < 4; ++nt) {
        v2f b = *(const v2f*)&Bt[buf][(nt * 16 + l15) * TP + k0 + (half << 1)];
        acc[nt] = wmma_f32(a, b, acc[nt]);
      }
    }
    __syncthreads();
    buf ^= 1;
  }

  // C/D layout: lane<16 -> M=r, N=l15 ; lane>=16 -> M=r+8.
#pragma unroll
  for (int nt = 0; nt < 4; ++nt) {
    int col = n0 + nt * 16 + l15;
#pragma unroll
    for (int r = 0; r < 8; ++r) {
      int row = m0 + wave * 16 + half * 8 + r;
      Y[(size_t)row * D2 + col] = acc[nt][r];
    }
  }
}

// ---------------------------------------------------------------------------
// Kernel 3: banded attention. One block = 16 queries [q0, q0+16), band
// columns j in [q0-64, q0+16) => 5 key tiles of 16.
// ---------------------------------------------------------------------------
#define QP  520  // LDS pitch for Q block rows (mult of 4)
#define SCP 84   // LDS pitch for 16x80 score/prob tile (even, mult of 4)

__global__ __launch_bounds__(128)
void attn_kernel(const float* __restrict__ Qm, const float* __restrict__ Km,
                 const float* __restrict__ V, float* __restrict__ retr,
                 float* __restrict__ attn) {
  __shared__ float qs[16 * QP];   // Q block  (16 x 512)
  __shared__ float sc[16 * SCP];  // scores -> probabilities (16 x 80)

  const int tid  = threadIdx.x;
  const int lane = tid & 31;
  const int wave = tid >> 5;
  const int half = lane >> 4;
  const int l15  = lane & 15;
  const int q0   = blockIdx.x * 16;
  const int b    = blockIdx.y;
  const int j0   = q0 - WIN;  // may be negative
  const size_t bbase = (size_t)b * SEQ * D2;
  const float* Qb = Qm + bbase;
  const float* Kb = Km + bbase;
  const float* Vb = V + bbase;

  // Stage Q block (16 x 512 fp32) into LDS via async-to-LDS.
  for (int idx = tid; idx < 16 * (D2 / 4); idx += 128) {
    int r  = idx >> 7;
    int c4 = (idx & 127) << 2;
    async_copy_b128(&Qb[(size_t)(q0 + r) * D2 + c4],
                    lds_off(&qs[r * QP + c4]));
  }
  asm volatile("s_wait_asynccnt 0x0" ::: "memory");
  __syncthreads();

  // --- Phase 1: scores S = Q K^T over the 5 band tiles (wave-uniform). ---
  for (int t = wave; t < 5; t += 4) {
    int jc0 = j0 + t * 16;
    int j   = jc0 + l15;
    int jcl = j < 0 ? 0 : j;  // clamp address; masked below
    const float* kp = &Kb[(size_t)jcl * D2 + (half << 1)];
    const float* ap = &qs[l15 * QP + (half << 1)];
    v8f acc = (v8f)0.0f;
#pragma unroll 8
    for (int d0 = 0; d0 < D2; d0 += 4) {
      v2f a  = *(const v2f*)&ap[d0];
      v2f bb = *(const v2f*)&kp[d0];
      acc = wmma_f32(a, bb, acc);
    }
#pragma unroll
    for (int r = 0; r < 8; ++r) {
      int m    = half * 8 + r;
      int i    = q0 + m;
      int diff = i - j;
      bool valid = (j >= 0) && (diff >= 1) && (diff <= WIN);
      sc[m * SCP + t * 16 + l15] = valid ? acc[r] * SCALE : -INFINITY;
    }
  }
  __syncthreads();

  // --- Phase 2: softmax over the 80-wide band (one thread per row). ---
  if (tid < 16) {
    float* row = &sc[tid * SCP];
    float mx = -INFINITY;
    for (int c = 0; c < BAND; ++c) mx = fmaxf(mx, row[c]);
    if (mx == -INFINITY) {  // row 0: no valid keys -> nan_to_num => zeros
      for (int c = 0; c < BAND; ++c) row[c] = 0.0f;
    } else {
      float s = 0.0f;
      for (int c = 0; c < BAND; ++c) { float e = __expf(row[c] - mx); row[c] = e; s += e; }
      float inv = 1.0f / s;
      for (int c = 0; c < BAND; ++c) row[c] *= inv;
    }
  }
  __syncthreads();

  // Scatter the normalized band into the dense attn output.
  for (int idx = tid; idx < 16 * BAND; idx += 128) {
    int m = idx / BAND;
    int c = idx - m * BAND;
    int j = j0 + c;
    if (j >= 0)
      attn[((size_t)b * SEQ + q0 + m) * SEQ + j] = sc[m * SCP + c];
  }

  // --- Phase 3: retrieved = P(16x80) @ V(80x512); wave owns 8 d-tiles. ---
#pragma unroll
  for (int dt = 0; dt < 8; ++dt) {
    const int dbase = (wave * 8 + dt) * 16;
    const float* ap = &sc[l15 * SCP + (half << 1)];
    v8f acc = (v8f)0.0f;
    for (int k0 = 0; k0 < BAND; k0 += 4) {
      v2f a  = *(const v2f*)&ap[k0];
      int jr  = j0 + k0 + (half << 1);
      int jr0 = jr < 0 ? 0 : jr;          // p==0 there, clamp address only
      int jr1 = (jr + 1) < 0 ? 0 : (jr + 1);
      v2f bb;
      bb.x = Vb[(size_t)jr0 * D2 + dbase + l15];
      bb.y = Vb[(size_t)jr1 * D2 + dbase + l15];
      acc = wmma_f32(a, bb, acc);
    }
#pragma unroll
    for (int r = 0; r < 8; ++r) {
      int row = q0 + half * 8 + r;
      retr[((size_t)b * SEQ + row) * D2 + dbase + l15] = acc[r];
    }
  }
}

// ---------------------------------------------------------------------------
extern "C" void kernel_launch(void* const* d_in, const int* in_sizes, int n_in,
                              void* d_out, int out_size, void* d_ws, size_t ws_size,
                              hipStream_t stream) {
  (void)in_sizes; (void)n_in; (void)out_size;

  const float* states = (const float*)d_in[0];  // [4,4096,256,2] == [16384,512]
  const float* Wq     = (const float*)d_in[1];  // [512,512]
  const float* Wk     = (const float*)d_in[2];  // [512,512]

  float* retr = (float*)d_out;                        // [4,4096,256,2]
  float* attn = retr + (size_t)NB * SEQ * D2;         // [4,4096,4096]

  // Workspace: Q then K, each 16384x512 fp32 (33.5 MB) -> 67 MB total.
  float* Qw = (float*)d_ws;
  float* Kw = Qw + (size_t)MROWS * D2;
  if (ws_size < (size_t)2 * MROWS * D2 * sizeof(float)) return;

  // 1) zero the 268 MB dense attn region (float4 grid-stride).
  size_t n4 = ((size_t)NB * SEQ * SEQ) / 4;
  zero_attn_kernel<<<4096, 256, 0, stream>>>((v4f*)attn, n4);

  // 2) Q and K projections (z selects weight/output pair).
  proj_kernel<<<dim3(D2 / 64, MROWS / 64, 2), 128, 0, stream>>>(states, Wq, Wk, Qw, Kw);

  // 3) banded attention + softmax + P@V.
  attn_kernel<<<dim3(SEQ / 16, NB), 128, 0, stream>>>(Qw, Kw, states, retr, attn);
}